// SSMBlock_76347338654363
// MI455X (gfx1250) — compile-verified
//
#include <hip/hip_runtime.h>

#define HIDDEN 512
#define BATCH  128
#define SEQ    512
#define ROWS   (BATCH * SEQ)

typedef float v2f __attribute__((ext_vector_type(2)));
typedef float v8f __attribute__((ext_vector_type(8)));

__device__ __forceinline__ v8f wmma4(v2f a, v2f b, v8f c) {
    // D = A(16x4 f32) * B(4x16 f32) + C(16x16 f32)
    return __builtin_amdgcn_wmma_f32_16x16x4_f32(
        /*neg_a=*/false, a, /*neg_b=*/false, b,
        /*c_mod=*/(short)0, c, /*reuse_a=*/false, /*reuse_b=*/false);
}

// ---------------------------------------------------------------------------
// Phase 1: U[r, n] = sum_k X[r, k] * B[k, n]
// Block = 512 threads (16 waves), tile 128(M) x 128(N), K-chunk 32,
// double-buffered LDS pipeline. B staged as K-pairs (float2) so each WMMA
// B-fragment is one ds_load_b64 with zero repacking.
// ---------------------------------------------------------------------------
#define P1_TM 128
#define P1_TN 128
#define P1_TK 32
#define P1_PK (P1_TK / 2)        // 16 k-pair rows per chunk
#define XS_STRIDE 36             // floats; all-64-bank conflict-free b64 reads
#define BS2_STRIDE 144           // float2 units; 2*144 mod 64 = 32 -> halves disjoint

__global__ __launch_bounds__(512) void ssm_phase1(const float* __restrict__ X,
                                                  const float* __restrict__ Bm,
                                                  float* __restrict__ U) {
    __shared__ float Xs[2][P1_TM * XS_STRIDE];            // 2 x 18 KB
    __shared__ float Bs[2][P1_PK * BS2_STRIDE * 2];       // 2 x 18 KB (float2 pairs)

    const int tid  = threadIdx.x;
    const int lane = tid & 31;
    const int wave = tid >> 5;
    const int half = lane >> 4;        // 0: lanes 0-15, 1: lanes 16-31
    const int l16  = lane & 15;
    const int bm   = blockIdx.x * P1_TM;
    const int bn   = blockIdx.y * P1_TN;
    const int wm   = (wave & 7) * 16;
    const int wn   = (wave >> 3) * 64;

    // staging coordinates (fixed per thread)
    const int bpk = tid >> 5;          // k-pair row 0..15
    const int bc4 = (tid & 31) << 2;   // col within N-tile, step 4

    v8f acc[4];
#pragma unroll
    for (int i = 0; i < 4; ++i)
#pragma unroll
        for (int j = 0; j < 8; ++j) acc[i][j] = 0.0f;

    float4 xr[2], br0, br1;

    auto load_global = [&](int k0) {
#pragma unroll
        for (int p = 0; p < 2; ++p) {
            const int f4  = tid + p * 512;
            const int row = f4 >> 3;
            const int c4  = (f4 & 7) << 2;
            xr[p] = *(const float4*)(X + (size_t)(bm + row) * HIDDEN + k0 + c4);
        }
        br0 = *(const float4*)(Bm + (size_t)(k0 + 2 * bpk)     * HIDDEN + bn + bc4);
        br1 = *(const float4*)(Bm + (size_t)(k0 + 2 * bpk + 1) * HIDDEN + bn + bc4);
    };

    auto store_lds = [&](int buf) {
#pragma unroll
        for (int p = 0; p < 2; ++p) {
            const int f4  = tid + p * 512;
            const int row = f4 >> 3;
            const int c4  = (f4 & 7) << 2;
            *(float4*)(&Xs[buf][row * XS_STRIDE + c4]) = xr[p];
        }
        v2f* bp = (v2f*)&Bs[buf][(bpk * BS2_STRIDE + bc4) * 2];
        bp[0] = (v2f){br0.x, br1.x};
        bp[1] = (v2f){br0.y, br1.y};
        bp[2] = (v2f){br0.z, br1.z};
        bp[3] = (v2f){br0.w, br1.w};
    };

    load_global(0);
    store_lds(0);
    __syncthreads();

    const int NCHUNK = HIDDEN / P1_TK;   // 16
    for (int c = 0; c < NCHUNK; ++c) {
        const int buf = c & 1;
        if (c + 1 < NCHUNK) load_global((c + 1) * P1_TK);   // overlap with compute

#pragma unroll
        for (int kk = 0; kk < P1_TK; kk += 4) {
            const int ak   = kk + half * 2;          // K pair base for this half-wave
            const int prow = (kk >> 1) + half;       // k-pair row in Bs
            v2f a = *(const v2f*)(&Xs[buf][(wm + l16) * XS_STRIDE + ak]);
#pragma unroll
            for (int i = 0; i < 4; ++i) {
                v2f b = *(const v2f*)(&Bs[buf][(prow * BS2_STRIDE + wn + i * 16 + l16) * 2]);
                acc[i] = wmma4(a, b, acc[i]);
            }
        }

        if (c + 1 < NCHUNK) store_lds(buf ^ 1);      // prior reads of buf^1 fenced by last sync
        __syncthreads();
    }

    // Store: VGPR j -> M = j + 8*half, lane -> N = l16
#pragma unroll
    for (int i = 0; i < 4; ++i)
#pragma unroll
        for (int j = 0; j < 8; ++j) {
            const int row = bm + wm + j + half * 8;
            const int col = bn + wn + i * 16 + l16;
            U[(size_t)row * HIDDEN + col] = acc[i][j];
        }
}

// ---------------------------------------------------------------------------
// Phase 2: sequential scan h_t = h_{t-1} @ A^T + u_t.
// 8 blocks x 1024 threads (32 waves); wave w owns N-tile [16w, 16w+16).
// h_{t-1} in LDS (stride 516 -> all 32 lanes distinct banks); A rows stream
// from L2/WGP$ as contiguous b64 loads; u_t read from d_out and overwritten
// in place with h_t. 4 interleaved accumulators cap the WMMA RAW chain at 32.
// ---------------------------------------------------------------------------
#define HS_STRIDE 516

__global__ __launch_bounds__(1024) void ssm_phase2(const float* __restrict__ h0,
                                                   const float* __restrict__ A,
                                                   float* __restrict__ UH) {
    __shared__ float Hs[16 * HS_STRIDE];   // ~33 KB

    const int tid  = threadIdx.x;
    const int lane = tid & 31;
    const int wave = tid >> 5;       // 0..31
    const int half = lane >> 4;
    const int l16  = lane & 15;
    const int m0   = blockIdx.x * 16;
    const int n0   = wave * 16;

    // Load h0 tile (16 x 512) into LDS
#pragma unroll
    for (int p = 0; p < 2; ++p) {
        const int f4  = tid + p * 1024;
        const int row = f4 >> 7;
        const int c4  = (f4 & 127) << 2;
        float4 v = *(const float4*)(h0 + (size_t)(m0 + row) * HIDDEN + c4);
        *(float4*)(&Hs[row * HS_STRIDE + c4]) = v;
    }
    __syncthreads();

    const float* Arow = A + (size_t)(n0 + l16) * HIDDEN;

    for (int t = 0; t < SEQ; ++t) {
        v8f acc[4];
        // Seed acc[0] with u_t (C-operand layout), zero the rest
#pragma unroll
        for (int j = 0; j < 8; ++j) {
            const int row = m0 + j + half * 8;
            acc[0][j] = UH[((size_t)row * SEQ + t) * HIDDEN + n0 + l16];
            acc[1][j] = 0.0f;
            acc[2][j] = 0.0f;
            acc[3][j] = 0.0f;
        }

#pragma unroll 4
        for (int kb = 0; kb < HIDDEN; kb += 16) {
#pragma unroll
            for (int i = 0; i < 4; ++i) {
                const int k = kb + i * 4 + half * 2;
                v2f a = *(const v2f*)(&Hs[l16 * HS_STRIDE + k]);  // h_{t-1}[M=l16, k..k+1]
                v2f b = *(const v2f*)(Arow + k);                  // A[n0+l16, k..k+1]
                acc[i] = wmma4(a, b, acc[i]);
            }
        }
        __syncthreads();   // all waves done reading h_{t-1}

#pragma unroll
        for (int j = 0; j < 8; ++j) {
            const float v = acc[0][j] + acc[1][j] + acc[2][j] + acc[3][j];
            const int row = j + half * 8;
            const int col = n0 + l16;
            Hs[row * HS_STRIDE + col] = v;
            UH[((size_t)(m0 + row) * SEQ + t) * HIDDEN + col] = v;
        }
        __syncthreads();   // h_t visible before next step
    }
}

extern "C" void kernel_launch(void* const* d_in, const int* in_sizes, int n_in,
                              void* d_out, int out_size, void* d_ws, size_t ws_size,
                              hipStream_t stream) {
    const float* h0 = (const float*)d_in[0];  // [128, 512]
    const float* x  = (const float*)d_in[1];  // [128, 512, 512]
    const float* A  = (const float*)d_in[2];  // [512, 512]
    const float* B  = (const float*)d_in[3];  // [512, 512]
    float* out = (float*)d_out;               // [128, 512, 512]

    dim3 g1(ROWS / P1_TM, HIDDEN / P1_TN);    // (512, 4)
    ssm_phase1<<<g1, 512, 0, stream>>>(x, B, out);

    ssm_phase2<<<BATCH / 16, 1024, 0, stream>>>(h0, A, out);
}